// exp_LL_td_2_2181843386536
// MI455X (gfx1250) — compile-verified
//
#include <hip/hip_runtime.h>
#include <hip/hip_bf16.h>

// ---------------------------------------------------------------------------
// exp_LL_td: per-sample 16x16 Lindbladian matrix exponential on CDNA5 WMMA.
//   DT*L[s] = P (256x30) @ [gamma(s); omega(s)]   (P precomputed on device)
//   E = expm(DT*L)  via scaling-squaring + Horner Taylor, 16x16 matmuls done
//   as 4x v_wmma_f32_16x16x4_f32 per matmul, one wave per sample.
// ---------------------------------------------------------------------------

#define NFREQ   199
#define DTC     0.05f
#define TPER    5.0f
#define NORMINV (1.0f / 150.0f)
#define TWO_PI  6.283185307179586f

typedef __attribute__((ext_vector_type(2))) float v2f;
typedef __attribute__((ext_vector_type(8))) float v8f;

static __device__ __forceinline__ v8f wmma_f32_16x16x4(v2f a, v2f b, v8f c) {
  // D(16x16 f32) = A(16x4 f32) x B(4x16 f32) + C
  return __builtin_amdgcn_wmma_f32_16x16x4_f32(false, a, false, b, (short)0, c,
                                               false, false);
}

// ---------------------------------------------------------------------------
// Setup 1: u = expm(i*theta) (15x15 complex, Hermitian theta), then
//          A_re/A_im[(i*15+l)*15+j] = Re/Im( u[i,j] * conj(u[l,j]) ).
// Single block of 256 threads; trivial cost.
// ---------------------------------------------------------------------------
__global__ __launch_bounds__(256) void k_expm_u(const float* __restrict__ u_re,
                                                const float* __restrict__ u_im,
                                                float* __restrict__ Are,
                                                float* __restrict__ Aim) {
  __shared__ float Xr[225], Xi[225], Yr[225], Yi[225], Tr[225], Ti[225];
  __shared__ int sc_sh;
  const int tid = threadIdx.x;
  if (tid < 225) {
    int i = tid / 15, j = tid % 15;
    float tr = u_re[i * 15 + j] + u_re[j * 15 + i];  // theta_re (symmetric)
    float ti = u_im[i * 15 + j] - u_im[j * 15 + i];  // theta_im (antisym)
    Xr[tid] = -ti;  // (i*theta)_re
    Xi[tid] = tr;   // (i*theta)_im
  }
  __syncthreads();
  if (tid == 0) {
    float nrm = 0.f;
    for (int i = 0; i < 15; ++i) {
      float rs = 0.f;
      for (int j = 0; j < 15; ++j)
        rs += fabsf(Xr[i * 15 + j]) + fabsf(Xi[i * 15 + j]);
      nrm = fmaxf(nrm, rs);
    }
    int sc = 0;
    if (nrm > 0.f) {
      int e = 0;
      frexpf(nrm, &e);
      sc = e + 1;
      if (sc < 0) sc = 0;
      if (sc > 30) sc = 30;
    }
    sc_sh = sc;
  }
  __syncthreads();
  const int sc = sc_sh;
  const float scl = ldexpf(1.0f, -sc);
  if (tid < 225) { Xr[tid] *= scl; Xi[tid] *= scl; }
  const int MORD = 18;
  __syncthreads();
  if (tid < 225) {
    int i = tid / 15, j = tid % 15;
    float id = (i == j) ? 1.f : 0.f;
    Yr[tid] = id + Xr[tid] * (1.0f / MORD);
    Yi[tid] = Xi[tid] * (1.0f / MORD);
  }
  for (int k = MORD - 1; k >= 1; --k) {
    __syncthreads();
    if (tid < 225) {
      int i = tid / 15, j = tid % 15;
      float ar = 0.f, ai = 0.f;
      for (int p = 0; p < 15; ++p) {
        float xr = Xr[i * 15 + p], xi = Xi[i * 15 + p];
        float yr = Yr[p * 15 + j], yi = Yi[p * 15 + j];
        ar += xr * yr - xi * yi;
        ai += xr * yi + xi * yr;
      }
      Tr[tid] = ar;
      Ti[tid] = ai;
    }
    __syncthreads();
    if (tid < 225) {
      int i = tid / 15, j = tid % 15;
      float id = (i == j) ? 1.f : 0.f;
      float inv = 1.0f / (float)k;
      Yr[tid] = id + Tr[tid] * inv;
      Yi[tid] = Ti[tid] * inv;
    }
  }
  for (int it = 0; it < sc; ++it) {
    __syncthreads();
    if (tid < 225) {
      int i = tid / 15, j = tid % 15;
      float ar = 0.f, ai = 0.f;
      for (int p = 0; p < 15; ++p) {
        float xr = Yr[i * 15 + p], xi = Yi[i * 15 + p];
        float yr = Yr[p * 15 + j], yi = Yi[p * 15 + j];
        ar += xr * yr - xi * yi;
        ai += xr * yi + xi * yr;
      }
      Tr[tid] = ar;
      Ti[tid] = ai;
    }
    __syncthreads();
    if (tid < 225) { Yr[tid] = Tr[tid]; Yi[tid] = Ti[tid]; }
  }
  __syncthreads();
  for (int idx = tid; idx < 15 * 15 * 15; idx += 256) {
    int i = idx / 225, rem = idx % 225, l = rem / 15, j = rem % 15;
    float uir = Yr[i * 15 + j], uii = Yi[i * 15 + j];
    float ulr = Yr[l * 15 + j], uli = Yi[l * 15 + j];
    Are[(i * 15 + l) * 15 + j] = uir * ulr + uii * uli;
    Aim[(i * 15 + l) * 15 + j] = uii * ulr - uir * uli;
  }
}

// ---------------------------------------------------------------------------
// Setup 2: P[e][q], e = row*16+col of the 16x16 L matrix, q in [0,30):
//   q<15  -> coefficient of gamma[q];  q>=15 -> coefficient of omega[q-15].
// Row 0 of L is zero. L[r][0] = tr_id, L[r][c] (r,c>=1) = h_comm + dsx.
// DT folded in. 7680 threads total, each does the k/i/j contraction.
// ---------------------------------------------------------------------------
__global__ __launch_bounds__(256) void k_build_P(const float* __restrict__ fS,
                                                 const float* __restrict__ dS,
                                                 const float* __restrict__ Are,
                                                 const float* __restrict__ Aim,
                                                 float* __restrict__ P) {
  int gid = blockIdx.x * 256 + threadIdx.x;
  if (gid >= 256 * 30) return;
  int e = gid / 30, q = gid % 30;
  int row = e / 16, col = e % 16;
  float val = 0.f;
  if (row >= 1) {
    int m = row - 1;
    if (col == 0) {
      if (q < 15) {  // tr_id[m] = 4 * sum_{i,j} f[i,m,j] * c_im[i,j]
        int g = q;
        float a = 0.f;
        for (int i = 0; i < 15; ++i)
          for (int j = 0; j < 15; ++j)
            a += fS[(i * 15 + m) * 15 + j] * Aim[(i * 15 + j) * 15 + g];
        val = 4.f * a;
      }
    } else {
      int n = col - 1;
      if (q < 15) {  // dsx gamma coefficient
        int g = q;
        float a = 0.f;
        for (int i = 0; i < 15; ++i) {
          for (int j = 0; j < 15; ++j) {
            float w1 = 0.f, w2 = 0.f;
            for (int k = 0; k < 15; ++k) {
              float fmjk = fS[(m * 15 + j) * 15 + k];
              float fnik = fS[(n * 15 + i) * 15 + k];
              float fmik = fS[(m * 15 + i) * 15 + k];
              float fnjk = fS[(n * 15 + j) * 15 + k];
              float dnik = dS[(n * 15 + i) * 15 + k];
              float dnjk = dS[(n * 15 + j) * 15 + k];
              w1 += fmjk * fnik + fmik * fnjk;   // -> c_re, weight -4
              w2 += fmjk * dnik - fmik * dnjk;   // -> c_im, weight +4
            }
            a += (-4.f * w1) * Are[(i * 15 + j) * 15 + g] +
                 (4.f * w2) * Aim[(i * 15 + j) * 15 + g];
          }
        }
        val = a;
      } else {  // h_comm omega coefficient: -4 f[k,m,n]
        int k = q - 15;
        val = -4.f * fS[(k * 15 + m) * 15 + n];
      }
    }
  }
  P[e * 30 + q] = val * DTC;
}

// ---------------------------------------------------------------------------
// Main kernel: one wave32 per sample; 8 waves (256 threads) per block.
// ---------------------------------------------------------------------------
#define LDS_PITCH 17  // 16x16 tile with +1 pad to dodge bank conflicts

__global__ __launch_bounds__(256) void k_main(
    const float* __restrict__ t, const float* __restrict__ x,
    const float* __restrict__ fg, const float* __restrict__ ag,
    const float* __restrict__ wg, const float* __restrict__ bg,
    const float* __restrict__ fo, const float* __restrict__ ao,
    const float* __restrict__ wo, const float* __restrict__ bo,
    const float* __restrict__ P, float* __restrict__ out, int B) {
  __shared__ float slab[8][16 * LDS_PITCH];
  const int wave = threadIdx.x >> 5;
  const int lane = threadIdx.x & 31;
  const int s = blockIdx.x * 8 + wave;
  if (s >= B) return;  // whole wave exits together
  float* L = slab[wave];

  // ---- Fourier features -> gamma, omega (lane-parallel + shfl reduce) ----
  const float tv = t[s];
  const float wph = TWO_PI * tv / TPER;
  float accg[15], acco[15];
#pragma unroll
  for (int d = 0; d < 15; ++d) { accg[d] = 0.f; acco[d] = 0.f; }
  for (int fi = lane; fi < NFREQ; fi += 32) {
    float sg, cg, so, co;
    sincosf(wph * fg[fi], &sg, &cg);
    sincosf(wph * fo[fi], &so, &co);
    float gc = ag[fi] * cg, gs = ag[NFREQ + fi] * sg;
    float oc = ao[fi] * co, os = ao[NFREQ + fi] * so;
#pragma unroll
    for (int d = 0; d < 15; ++d) {
      accg[d] += gc * wg[d * (2 * NFREQ) + fi] + gs * wg[d * (2 * NFREQ) + NFREQ + fi];
      acco[d] += oc * wo[d * (2 * NFREQ) + fi] + os * wo[d * (2 * NFREQ) + NFREQ + fi];
    }
  }
#pragma unroll
  for (int d = 0; d < 15; ++d) {
#pragma unroll
    for (int m = 1; m < 32; m <<= 1) {
      accg[d] += __shfl_xor(accg[d], m, 32);
      acco[d] += __shfl_xor(acco[d], m, 32);
    }
  }
  float gam[15], om[15];
#pragma unroll
  for (int d = 0; d < 15; ++d) {
    float g = accg[d] + bg[d];
    gam[d] = g * g * NORMINV;
    om[d] = (acco[d] + bo[d]) * NORMINV;
  }

  // ---- Build DT*L directly in WMMA C/D fragment layout ----
  const int col = lane & 15;
  const int roff = (lane & 16) ? 8 : 0;
  v8f F;
#pragma unroll
  for (int r = 0; r < 8; ++r) {
    int row = r + roff;
    const float* p = &P[(row * 16 + col) * 30];
    float a = 0.f;
#pragma unroll
    for (int g = 0; g < 15; ++g) a += p[g] * gam[g];
#pragma unroll
    for (int k = 0; k < 15; ++k) a += p[15 + k] * om[k];
    F[r] = a;
  }

  // ---- inf-norm via shuffles, pick scaling s with ||X/2^s|| < 0.5 ----
  float nrm = 0.f;
#pragma unroll
  for (int r = 0; r < 8; ++r) {
    float v = fabsf(F[r]);
    v += __shfl_xor(v, 1, 32);
    v += __shfl_xor(v, 2, 32);
    v += __shfl_xor(v, 4, 32);
    v += __shfl_xor(v, 8, 32);
    nrm = fmaxf(nrm, v);
  }
  nrm = fmaxf(nrm, __shfl_xor(nrm, 16, 32));
  int sc = 0;
  if (nrm > 0.f) {
    int e = 0;
    frexpf(nrm, &e);
    sc = e + 1;
    if (sc < 0) sc = 0;
    if (sc > 24) sc = 24;
  }
  const float scl = ldexpf(1.0f, -sc);
#pragma unroll
  for (int r = 0; r < 8; ++r) F[r] *= scl;

  // ---- X to LDS; build fixed A-fragments of X (4 K-slices) ----
#pragma unroll
  for (int r = 0; r < 8; ++r) L[(r + roff) * LDS_PITCH + col] = F[r];
  const int am = lane & 15;            // A row
  const int ksub = (lane & 16) ? 2 : 0;  // K sub-offset for this lane half
  v2f aF[4];
#pragma unroll
  for (int kk = 0; kk < 4; ++kk) {
    int kb = 4 * kk + ksub;
    aF[kk].x = L[am * LDS_PITCH + kb];
    aF[kk].y = L[am * LDS_PITCH + kb + 1];
  }

  // ---- Horner Taylor: Y = I + X/M; Y = I + (X*Y)/k ----
  const int MORD = 10;
  v8f Y;
#pragma unroll
  for (int r = 0; r < 8; ++r)
    Y[r] = ((r + roff) == col ? 1.f : 0.f) + F[r] * (1.0f / MORD);
  for (int k = MORD - 1; k >= 1; --k) {
    // Y -> LDS, read back B-fragments; D = X * Y
#pragma unroll
    for (int r = 0; r < 8; ++r) L[(r + roff) * LDS_PITCH + col] = Y[r];
    v8f acc = {0.f, 0.f, 0.f, 0.f, 0.f, 0.f, 0.f, 0.f};
#pragma unroll
    for (int kk = 0; kk < 4; ++kk) {
      v2f bF;
      int kb = 4 * kk + ksub;
      bF.x = L[(kb + 0) * LDS_PITCH + col];
      bF.y = L[(kb + 1) * LDS_PITCH + col];
      acc = wmma_f32_16x16x4(aF[kk], bF, acc);
    }
    float inv = 1.0f / (float)k;
#pragma unroll
    for (int r = 0; r < 8; ++r)
      Y[r] = ((r + roff) == col ? 1.f : 0.f) + acc[r] * inv;
  }

  // ---- sc squarings: Y = Y*Y ----
  for (int it = 0; it < sc; ++it) {
#pragma unroll
    for (int r = 0; r < 8; ++r) L[(r + roff) * LDS_PITCH + col] = Y[r];
    v8f acc = {0.f, 0.f, 0.f, 0.f, 0.f, 0.f, 0.f, 0.f};
#pragma unroll
    for (int kk = 0; kk < 4; ++kk) {
      int kb = 4 * kk + ksub;
      v2f a2, b2;
      a2.x = L[am * LDS_PITCH + kb];
      a2.y = L[am * LDS_PITCH + kb + 1];
      b2.x = L[(kb + 0) * LDS_PITCH + col];
      b2.y = L[(kb + 1) * LDS_PITCH + col];
      acc = wmma_f32_16x16x4(a2, b2, acc);
    }
    Y = acc;
  }

  // ---- readout: out[s,j] = E[j+1][0] + sum_i x[s,i] * E[j+1][i+1] ----
#pragma unroll
  for (int r = 0; r < 8; ++r) L[(r + roff) * LDS_PITCH + col] = Y[r];
  if (lane < 15) {
    const int j = lane;
    float o = L[(j + 1) * LDS_PITCH + 0];
    const float* xs = x + (size_t)s * 15;
#pragma unroll
    for (int i = 0; i < 15; ++i) o += xs[i] * L[(j + 1) * LDS_PITCH + (i + 1)];
    out[(size_t)s * 15 + j] = o;
  }
}

// ---------------------------------------------------------------------------
extern "C" void kernel_launch(void* const* d_in, const int* in_sizes, int n_in,
                              void* d_out, int out_size, void* d_ws,
                              size_t ws_size, hipStream_t stream) {
  const float* t    = (const float*)d_in[0];
  const float* x    = (const float*)d_in[1];
  const float* u_re = (const float*)d_in[2];
  const float* u_im = (const float*)d_in[3];
  const float* fg   = (const float*)d_in[4];
  const float* ag   = (const float*)d_in[5];
  const float* wg   = (const float*)d_in[6];
  const float* bg   = (const float*)d_in[7];
  const float* fo   = (const float*)d_in[8];
  const float* ao   = (const float*)d_in[9];
  const float* wo   = (const float*)d_in[10];
  const float* bo   = (const float*)d_in[11];
  const float* fS   = (const float*)d_in[12];
  const float* dS   = (const float*)d_in[13];

  float* ws  = (float*)d_ws;
  float* P   = ws;                 // 256*30 = 7680 floats
  float* Are = ws + 7680;          // 3375 floats
  float* Aim = ws + 7680 + 3375;   // 3375 floats

  const int B = in_sizes[0];  // t is (B,1)

  k_expm_u<<<1, 256, 0, stream>>>(u_re, u_im, Are, Aim);
  k_build_P<<<30, 256, 0, stream>>>(fS, dS, Are, Aim, P);
  k_main<<<(B + 7) / 8, 256, 0, stream>>>(t, x, fg, ag, wg, bg, fo, ao, wo, bo,
                                          P, (float*)d_out, B);
}